// SC_8856222564940
// MI455X (gfx1250) — compile-verified
//
#include <hip/hip_runtime.h>
#include <stdint.h>
#include <math.h>

#define NN 2048          // row length (reference N)
#define CHUNKS 16        // j-chunks per batch in rowsum kernel
#define CHUNK 128        // j's per block (CHUNKS*CHUNK == NN)

typedef unsigned int u32;
typedef u32   v4u __attribute__((ext_vector_type(4)));
typedef int   v4i __attribute__((ext_vector_type(4)));
typedef int   v8i __attribute__((ext_vector_type(8)));
typedef float v4f __attribute__((ext_vector_type(4)));   // native vector (NT-store capable)

#if defined(__AMDGCN__) && __has_builtin(__builtin_amdgcn_tensor_load_to_lds)
#define HAVE_TDM 1
#else
#define HAVE_TDM 0
#endif

#if defined(__AMDGCN__) && __has_builtin(__builtin_amdgcn_exp2f)
#define FAST_EXP2(x) __builtin_amdgcn_exp2f(x)
#else
#define FAST_EXP2(x) exp2f(x)
#endif

#if HAVE_TDM
// Issue one Tensor-Data-Mover load of NN contiguous f32 (one batch row) into LDS.
// D# built per CDNA5 ISA ch. 8 (group0: count/lds_addr/global_addr/type=2;
// group1: data_size=4B, tensor_dim0=NN, tensor_dim1=1, tile_dim0=NN, stride=NN).
__device__ __forceinline__ void tdm_load_row_to_lds(const float* gsrc, u32 lds_byte_off) {
  uint64_t ga = (uint64_t)(uintptr_t)gsrc;
  v4u g0 = {
    1u,                                            // count=1, is_restore=0, gather off
    lds_byte_off,                                  // lds_addr (bytes)
    (u32)(ga & 0xFFFFFFFFu),                       // global_addr[31:0]
    ((u32)((ga >> 32) & 0x01FFFFFFu)) | (2u << 30) // global_addr[56:32] | type=2
  };
  v8i g1 = {
    (int)(2u << 16),                        // wg_mask=0 | data_size=2 (4 bytes)
    (int)(((u32)NN & 0xFFFFu) << 16),       // atomic_barrier=0 | tensor_dim0[15:0]
    (int)(((u32)NN >> 16) | (1u << 16)),    // tensor_dim0[31:16] | tensor_dim1[15:0]=1
    (int)(((u32)NN) << 16),                 // tensor_dim1[31:16]=0 | tile_dim0=NN
    0,                                      // tile_dim1=0 (unused), tile_dim2=0
    (int)NN,                                // tensor_dim0_stride[31:0]
    0,                                      // stride0 hi | tensor_dim1_stride lo
    0
  };
  v4i z4 = {0, 0, 0, 0};
#if __has_include(<hip/amd_detail/amd_gfx1250_TDM.h>)
  v8i z8 = {0, 0, 0, 0, 0, 0, 0, 0};
  __builtin_amdgcn_tensor_load_to_lds(g0, g1, z4, z4, z8, 0);   // clang-23 / therock arity
#else
  __builtin_amdgcn_tensor_load_to_lds(g0, g1, z4, z4, 0);       // ROCm 7.2 arity
#endif
}
#endif

// ---------------- Kernel 1: rowsum[b,j] = sum_l |s[b,j] - s[b,l]| -------------
__global__ __launch_bounds__(CHUNK) void SC_neuralsort_rowsum(
    const float* __restrict__ scores, float* __restrict__ rowsum) {
  __shared__ __align__(16) float s[NN];
  const int b     = (int)blockIdx.x >> 4;   // blockIdx.x / CHUNKS
  const int chunk = (int)blockIdx.x & (CHUNKS - 1);
  const float* row = scores + (size_t)b * NN;

#if HAVE_TDM
  if (threadIdx.x < 32u) {                  // wave 0 only issues the TDM op
    tdm_load_row_to_lds(row, (u32)(uintptr_t)(&s[0]));
    __builtin_amdgcn_s_wait_tensorcnt(0);
  }
  __syncthreads();
#else
  for (int i = threadIdx.x; i < NN; i += (int)blockDim.x) s[i] = row[i];
  __syncthreads();
#endif

  const int j    = chunk * CHUNK + (int)threadIdx.x;
  const float sj = s[j];
  float acc = 0.0f;
#pragma unroll 4
  for (int l = 0; l < NN; l += 4) {
    v4f v = *reinterpret_cast<const v4f*>(&s[l]);   // ds_load_b128, conflict-free broadcast
    acc += fabsf(sj - v.x);
    acc += fabsf(sj - v.y);
    acc += fabsf(sj - v.z);
    acc += fabsf(sj - v.w);
  }
  rowsum[(size_t)b * NN + j] = acc;
}

// ---------------- Kernel 2: P_hat[b,i,:] = softmax((scale_i*s - rowsum)/tau) --
__device__ __forceinline__ float wave_max32(float v) {
#pragma unroll
  for (int off = 16; off > 0; off >>= 1) v = fmaxf(v, __shfl_xor(v, off, 32));
  return v;
}
__device__ __forceinline__ float wave_sum32(float v) {
#pragma unroll
  for (int off = 16; off > 0; off >>= 1) v += __shfl_xor(v, off, 32);
  return v;
}

__global__ __launch_bounds__(256) void SC_neuralsort_softmax(
    const float* __restrict__ scores, const float* __restrict__ rowsum,
    const float* __restrict__ tau_p, float* __restrict__ out, int K) {
  const int b = (int)blockIdx.x / K;
  const int i = (int)blockIdx.x - b * K;

  const float* srow = scores + (size_t)b * NN;
  const float* rrow = rowsum + (size_t)b * NN;
  float*       orow = out + ((size_t)b * K + i) * NN;

  // Prefetch the two 8KB rows (global_prefetch_b8); they are L2-hot but this
  // hides WGP$ fill latency for the first vector loads.
  __builtin_prefetch(srow + threadIdx.x * 8, 0, 3);
  __builtin_prefetch(rrow + threadIdx.x * 8, 0, 3);

  // Fold 1/tau and log2(e) into the logit scale once: with c = log2e/tau > 0,
  // z' = c*(scale_i*s - rowsum) preserves the argmax, and exp((z-m)/tau) ==
  // exp2(z' - m'). Saves one VALU mul per element in the exp loop.
  const float LOG2E = 1.4426950408889634f;
  const float c       = LOG2E / tau_p[0];
  const float scale_c = (float)(NN + 1 - 2 * (i + 1)) * c;  // scaling[i] * c
  const float neg_c   = -c;

  const int wave = (int)threadIdx.x >> 5;
  const int lane = (int)threadIdx.x & 31;
  __shared__ float red_max[8];
  __shared__ float red_sum[8];

  // Each thread owns 2 float4 groups: j = 4*(tid + g*256) .. +3  (covers 2048)
  float z[8];
  float lmax = -INFINITY;
#pragma unroll
  for (int g = 0; g < 2; ++g) {
    const int j4 = ((int)threadIdx.x + g * 256) * 4;
    v4f sv = *reinterpret_cast<const v4f*>(srow + j4);
    v4f rv = *reinterpret_cast<const v4f*>(rrow + j4);
    float* zz = &z[g * 4];
    zz[0] = fmaf(scale_c, sv.x, neg_c * rv.x);
    zz[1] = fmaf(scale_c, sv.y, neg_c * rv.y);
    zz[2] = fmaf(scale_c, sv.z, neg_c * rv.z);
    zz[3] = fmaf(scale_c, sv.w, neg_c * rv.w);
    lmax = fmaxf(lmax, fmaxf(fmaxf(zz[0], zz[1]), fmaxf(zz[2], zz[3])));
  }

  // Block max (wave32 shuffle tree + 8-wave LDS combine)
  float m = wave_max32(lmax);
  if (lane == 0) red_max[wave] = m;
  __syncthreads();
  if (wave == 0) {
    float t = (lane < 8) ? red_max[lane] : -INFINITY;
    t = wave_max32(t);
    if (lane == 0) red_max[0] = t;
  }
  __syncthreads();
  m = red_max[0];

  // exp2 and block sum
  float p[8];
  float lsum = 0.0f;
#pragma unroll
  for (int q = 0; q < 8; ++q) {
    p[q] = FAST_EXP2(z[q] - m);   // v_exp_f32; z-m <= 0, large-negative -> 0
    lsum += p[q];
  }
  float ssum = wave_sum32(lsum);
  if (lane == 0) red_sum[wave] = ssum;
  __syncthreads();
  if (wave == 0) {
    float t = (lane < 8) ? red_sum[lane] : 0.0f;
    t = wave_sum32(t);
    if (lane == 0) red_sum[0] = t;
  }
  __syncthreads();
  const float inv_sum = 1.0f / red_sum[0];

#pragma unroll
  for (int g = 0; g < 2; ++g) {
    const int j4 = ((int)threadIdx.x + g * 256) * 4;
    v4f o;
    o.x = p[g * 4 + 0] * inv_sum;
    o.y = p[g * 4 + 1] * inv_sum;
    o.z = p[g * 4 + 2] * inv_sum;
    o.w = p[g * 4 + 3] * inv_sum;
    // Output is write-once, never re-read by these kernels: non-temporal store.
    __builtin_nontemporal_store(o, reinterpret_cast<v4f*>(orow + j4));
  }
}

extern "C" void kernel_launch(void* const* d_in, const int* in_sizes, int n_in,
                              void* d_out, int out_size, void* d_ws, size_t ws_size,
                              hipStream_t stream) {
  const float* scores = (const float*)d_in[0];   // [B, N] f32
  // d_in[1] = k (int, value known via out_size), d_in[2] = tau (f32 scalar array)
  const float* tau = (const float*)d_in[2];

  const int total = in_sizes[0];        // B * N
  const int B     = total / NN;         // 16
  const int K     = out_size / total;   // 10  (out = B*K*N)

  float* rowsum_buf = (float*)d_ws;     // B*N f32 = 128 KB scratch

  SC_neuralsort_rowsum<<<B * CHUNKS, CHUNK, 0, stream>>>(scores, rowsum_buf);
  SC_neuralsort_softmax<<<B * K, 256, 0, stream>>>(scores, rowsum_buf, tau,
                                                   (float*)d_out, K);
}